// KalmanNet_38182259261799
// MI455X (gfx1250) — compile-verified
//
#include <hip/hip_runtime.h>

#define B_TOT    512
#define T_STEPS  256
#define HID      400
#define H3       1200
#define BC       16            // batch rows per block (WMMA N)
#define NBLK     (B_TOT / BC)  // 32 persistent blocks
#define NTHREADS 256           // 8 wave32
#define NWAVES   8
#define TILES_PER_WAVE 10      // ceil(75/8) padded to 80 tiles

typedef float v2f __attribute__((ext_vector_type(2)));
typedef float v8f __attribute__((ext_vector_type(8)));

__device__ __forceinline__ float fast_sigmoid(float x) {
    return 1.0f / (1.0f + __expf(-x));
}
__device__ __forceinline__ float fast_tanh(float x) {
    x = fminf(15.0f, fmaxf(-15.0f, x));
    float e = __expf(2.0f * x);
    return (e - 1.0f) / (e + 1.0f);
}

// ---------------------------------------------------------------------------
// Fold the input path:  W_eff = W_ih @ Wi  [1200,6],  b_eff = W_ih @ bi + b_ih
// (valid because there is no nonlinearity between the Wi and W_ih layers)
// ---------------------------------------------------------------------------
__global__ void knet_fold_kernel(const float* __restrict__ W_ih,
                                 const float* __restrict__ Wi,
                                 const float* __restrict__ bi,
                                 const float* __restrict__ b_ih,
                                 float* __restrict__ Weff,
                                 float* __restrict__ beff) {
    int j = blockIdx.x * blockDim.x + threadIdx.x;
    if (j >= H3) return;
    float a0 = 0.f, a1 = 0.f, a2 = 0.f, a3 = 0.f, a4 = 0.f, a5 = 0.f, ab = 0.f;
    const float* wr = W_ih + (size_t)j * HID;
    for (int k = 0; k < HID; ++k) {
        float w = wr[k];
        const float* wik = Wi + k * 6;
        a0 += w * wik[0]; a1 += w * wik[1]; a2 += w * wik[2];
        a3 += w * wik[3]; a4 += w * wik[4]; a5 += w * wik[5];
        ab += w * bi[k];
    }
    float* o = Weff + j * 6;
    o[0] = a0; o[1] = a1; o[2] = a2; o[3] = a3; o[4] = a4; o[5] = a5;
    beff[j] = ab + b_ih[j];
}

// ---------------------------------------------------------------------------
// Persistent recurrent kernel: each block owns 16 batch rows for all 256 steps.
// h state lives in LDS; gh = W_hh @ h done with V_WMMA_F32_16X16X4_F32.
// ---------------------------------------------------------------------------
__global__ __launch_bounds__(NTHREADS) void knet_step_kernel(
    const float* __restrict__ y_seq,  // [512,256,2]
    const float* __restrict__ W_hh,   // [1200,400]
    const float* __restrict__ b_hh,   // [1200]
    const float* __restrict__ Wo,     // [8,400]
    const float* __restrict__ bo,     // [8]
    const float* __restrict__ Weff,   // [1200,6]
    const float* __restrict__ beff,   // [1200]
    float* __restrict__ out)          // [512,256,4]
{
    extern __shared__ float smem[];
    // h stored k-pair-major: hp[(p*16+b)*2 + j] = h[k=2p+j][b]   (6400 floats)
    float* hp   = smem;
    float* grz  = hp   + 6400;   // combined r/z preact [800][16]  (12800)
    float* hn   = grz  + 12800;  // h-side n preact     [400][16]  (6400)
    float* nn   = hn   + 6400;   // GRU input           [6][16]    (96)
    float* xpri = nn   + 96;     // x_pri               [16][4]    (64)
    float* inn  = xpri + 64;     // innovation          [16][2]    (32)
    float* xst  = inn  + 32;     // x_post state        [16][4]    (64)
    float* dlt  = xst  + 64;     // delta (prev corr)   [16][4]    (64)
    float* kmat = dlt  + 64;     // K gains             [16][8]    (128)

    const int tid  = threadIdx.x;
    const int b0   = blockIdx.x * BC;
    const int lane = tid & 31;
    const int wave = tid >> 5;
    const int nlo  = lane & 15;   // N index (batch) / M index for A rows
    const int h2   = lane >> 4;   // lane-half selector

    // zero-init recurrent state
    for (int i = tid; i < 6400; i += NTHREADS) hp[i] = 0.f;
    if (tid < 64) { xst[tid] = 0.f; dlt[tid] = 0.f; }
    __syncthreads();

    for (int t = 0; t < T_STEPS; ++t) {
        // ---- stage 1: per-batch dynamics + innovation (threads 0..15) ----
        if (tid < BC) {
            int b = tid;
            #pragma unroll
            for (int s = 0; s < 4; ++s) {
                float xv  = xst[b * 4 + s];
                float pri = xv + 0.1f * fast_tanh(xv);
                xpri[b * 4 + s] = pri;
                nn[s * 16 + b]  = dlt[b * 4 + s];
            }
            #pragma unroll
            for (int o = 0; o < 2; ++o) {
                float iv = y_seq[((size_t)(b0 + b) * T_STEPS + t) * 2 + o]
                           - xpri[b * 4 + o];
                inn[b * 2 + o]       = iv;
                nn[(4 + o) * 16 + b] = iv;
            }
        }
        __syncthreads();

        // ---- stage 2: gh = W_hh @ h via fp32 WMMA; fold gi for r/z rows ----
        const v2f* hpv = (const v2f*)hp;
        for (int pr = 0; pr < TILES_PER_WAVE; pr += 2) {
            const int t0 = wave * TILES_PER_WAVE + pr;
            const int t1 = t0 + 1;
            int m0 = t0 * 16 + nlo; if (m0 > H3 - 1) m0 = H3 - 1;  // pad clamp
            int m1 = t1 * 16 + nlo; if (m1 > H3 - 1) m1 = H3 - 1;
            const float* a0p = W_hh + (size_t)m0 * HID;
            const float* a1p = W_hh + (size_t)m1 * HID;
            v8f acc0 = {};
            v8f acc1 = {};
            #pragma unroll 4
            for (int s = 0; s < HID / 4; ++s) {
                int kp = 2 * s + h2;                         // float-pair index
                v2f bfrag  = hpv[kp * 16 + nlo];             // h[2kp..2kp+1][n]
                v2f afrag0 = *(const v2f*)(a0p + 2 * kp);    // W_hh[m][2kp..+1]
                v2f afrag1 = *(const v2f*)(a1p + 2 * kp);
                acc0 = __builtin_amdgcn_wmma_f32_16x16x4_f32(
                    false, afrag0, false, bfrag, (short)0, acc0, false, false);
                acc1 = __builtin_amdgcn_wmma_f32_16x16x4_f32(
                    false, afrag1, false, bfrag, (short)0, acc1, false, false);
            }
            #pragma unroll
            for (int which = 0; which < 2; ++which) {
                int tt    = which ? t1 : t0;
                v8f acc   = which ? acc1 : acc0;
                int mbase = tt * 16;
                if (mbase < H3) {                      // uniform per wave
                    if (mbase < 800) {                 // r/z region (tiles 0..49)
                        #pragma unroll
                        for (int r = 0; r < 8; ++r) {
                            int m = mbase + r + 8 * h2;       // C/D row map
                            float gi = beff[m];
                            const float* we = Weff + m * 6;
                            #pragma unroll
                            for (int c = 0; c < 6; ++c)
                                gi += we[c] * nn[c * 16 + nlo];
                            grz[m * 16 + nlo] = acc[r] + b_hh[m] + gi;
                        }
                    } else {                           // n region (tiles 50..74)
                        #pragma unroll
                        for (int r = 0; r < 8; ++r) {
                            int m = mbase + r + 8 * h2;
                            hn[(m - 800) * 16 + nlo] = acc[r] + b_hh[m];
                        }
                    }
                }
            }
        }
        __syncthreads();

        // ---- stage 3: gates + h update (gi_n computed inline, 6-wide) ----
        #pragma unroll 1
        for (int it = 0; it < (HID * BC) / NTHREADS; ++it) {   // 25 iters
            int idx = it * NTHREADS + tid;
            int j = idx >> 4, b = idx & 15;
            float rg = fast_sigmoid(grz[j * 16 + b]);
            float zg = fast_sigmoid(grz[(j + 400) * 16 + b]);
            float gi = beff[800 + j];
            const float* we = Weff + (800 + j) * 6;
            #pragma unroll
            for (int c = 0; c < 6; ++c) gi += we[c] * nn[c * 16 + b];
            float ng   = fast_tanh(gi + rg * hn[j * 16 + b]);
            int hidx   = ((j >> 1) * 16 + b) * 2 + (j & 1);
            float hold = hp[hidx];
            hp[hidx]   = (1.f - zg) * ng + zg * hold;
        }
        __syncthreads();

        // ---- stage 4: K = Wo @ h_new + bo  (threads 0..127) ----
        if (tid < BC * 8) {
            int o = tid & 7, b = tid >> 3;
            float acc = bo[o];
            const float* wo = Wo + o * HID;
            #pragma unroll 4
            for (int k = 0; k < HID; ++k)
                acc += wo[k] * hp[((k >> 1) * 16 + b) * 2 + (k & 1)];
            kmat[b * 8 + o] = acc;
        }
        __syncthreads();

        // ---- stage 5: Kalman correction, output, state update ----
        if (tid < BC * 4) {
            int s = tid & 3, b = tid >> 2;
            float corr = kmat[b * 8 + s * 2]     * inn[b * 2]
                       + kmat[b * 8 + s * 2 + 1] * inn[b * 2 + 1];
            float xp = xpri[b * 4 + s] + corr;
            out[((size_t)(b0 + b) * T_STEPS + t) * 4 + s] = xp;
            xst[b * 4 + s] = xp;
            dlt[b * 4 + s] = corr;
        }
        __syncthreads();
    }
}

// ---------------------------------------------------------------------------
extern "C" void kernel_launch(void* const* d_in, const int* in_sizes, int n_in,
                              void* d_out, int out_size, void* d_ws, size_t ws_size,
                              hipStream_t stream) {
    (void)in_sizes; (void)n_in; (void)out_size; (void)ws_size;
    const float* y_seq = (const float*)d_in[0];
    const float* Wi    = (const float*)d_in[1];
    const float* bi    = (const float*)d_in[2];
    const float* W_ih  = (const float*)d_in[3];
    const float* W_hh  = (const float*)d_in[4];
    const float* b_ih  = (const float*)d_in[5];
    const float* b_hh  = (const float*)d_in[6];
    const float* Wo    = (const float*)d_in[7];
    const float* bo    = (const float*)d_in[8];

    float* ws   = (float*)d_ws;
    float* Weff = ws;          // 1200*6 = 7200 floats
    float* beff = ws + 7200;   // 1200 floats

    knet_fold_kernel<<<(H3 + 255) / 256, 256, 0, stream>>>(
        W_ih, Wi, bi, b_ih, Weff, beff);

    const size_t lds_floats = 6400 + 12800 + 6400 + 96 + 64 + 32 + 64 + 64 + 128;
    knet_step_kernel<<<NBLK, NTHREADS, lds_floats * sizeof(float), stream>>>(
        y_seq, W_hh, b_hh, Wo, bo, Weff, beff, (float*)d_out);
}